// GeometryAttentionBias_59803124629519
// MI455X (gfx1250) — compile-verified
//
#include <hip/hip_runtime.h>

typedef __attribute__((ext_vector_type(2))) float v2f;
typedef __attribute__((ext_vector_type(8))) float v8f;

#define TILES_J_PER_WG 8   // 8 waves/block, each wave owns one 16x16 tile

// out[b][i][j] = exp(-(c_i-c_j)^2/(2 g^2)) + (n_i.n_j + 1)/2
//               - sigma*(d_i+d_j) + l_i*l_j - aniso[b][i][j]
//
// WMMA f32 16x16x4 computes per tile:  D = A(16x4) x B(4x16)
//   A row i = [0.5*nx_i, 0.5*ny_i, 0.5*nz_i, l_i]
//   B col j = [nx_j,     ny_j,     nz_j,     l_j]^T
//   => D = 0.5*(n_i.n_j) + l_i*l_j   (add 0.5 elementwise for norm_cons)
//
// aniso/out are a single-touch 512 MB stream: use non-temporal TH policy so
// the 192 MB L2 keeps only the small per-point arrays (hit by all 262k waves).
__global__ __launch_bounds__(256) void GeometryAttentionBias_kernel(
    const float* __restrict__ curvature,   // [B,N]
    const float* __restrict__ density,     // [B,N]
    const float* __restrict__ normals,     // [B,N,3]
    const float* __restrict__ linearity,   // [B,N]
    const float* __restrict__ aniso,       // [B,N,N]
    const float* __restrict__ gamma_p,     // [1]
    const float* __restrict__ sigma_p,     // [1]
    const float* __restrict__ cscale_p,    // [1]
    float* __restrict__ out,               // [B,N,N]
    int n)
{
    const int lane = threadIdx.x & 31;
    const int wave = threadIdx.x >> 5;

    const int tj = blockIdx.x * TILES_J_PER_WG + wave;
    const int ti = blockIdx.y;
    const int b  = blockIdx.z;

    const int i_base = ti * 16;
    const int j_base = tj * 16;

    const float gamma  = gamma_p[0];
    const float sigma  = sigma_p[0];
    const float cscale = cscale_p[0];
    const float inv2g2 = 1.0f / (2.0f * gamma * gamma);

    const int hi = lane >> 4;      // lane half: selects K pair (A/B) and M+8 (C/D)
    const int lr = lane & 15;

    const float* curB = curvature + (size_t)b * n;
    const float* denB = density   + (size_t)b * n;
    const float* linB = linearity + (size_t)b * n;
    const float* nrmB = normals   + (size_t)b * n * 3;

    // ---- A fragment: 16x4 f32.  VGPR0 = K0|K2, VGPR1 = K1|K3, M = lane&15 ----
    const int ia = i_base + lr;
    const float* na = nrmB + (size_t)ia * 3;
    v2f afrag;
    afrag.x = 0.5f * (hi ? na[2] : na[0]);               // K0: 0.5*nx | K2: 0.5*nz
    afrag.y = hi ? linB[ia] : 0.5f * na[1];              // K1: 0.5*ny | K3: l_i

    // ---- B fragment: 4x16 f32.  VGPR0 = K0|K2, VGPR1 = K1|K3, col = lane&15 ----
    const int jb = j_base + lr;
    const float* nb = nrmB + (size_t)jb * 3;
    v2f bfrag;
    bfrag.x = hi ? nb[2] : nb[0];                        // K0: nx | K2: nz
    bfrag.y = hi ? linB[jb] : nb[1];                     // K1: ny | K3: l_j

    v8f acc = {};
    // D = A x B + 0 : emits v_wmma_f32_16x16x4_f32
    acc = __builtin_amdgcn_wmma_f32_16x16x4_f32(
        /*neg_a=*/false, afrag, /*neg_b=*/false, bfrag,
        /*c_mod=*/(short)0, acc, /*reuse_a=*/false, /*reuse_b=*/false);

    // ---- Elementwise terms + fused non-temporal stream of aniso -> out ----
    // C/D layout: VGPR v holds (M = v + 8*hi, N = lane&15)
    const int j  = j_base + lr;
    const float cj = curB[j] * cscale;
    const float dj = denB[j];

    const size_t row0 = ((size_t)b * n + (i_base + hi * 8)) * n + j;
    const float* aRow = aniso + row0;
    float*       oRow = out   + row0;

#pragma unroll
    for (int v = 0; v < 8; ++v) {
        const int i  = i_base + hi * 8 + v;
        const float ci = curB[i] * cscale;
        const float di = denB[i];
        const float d  = ci - cj;
        const float cur_sim = __expf(-(d * d) * inv2g2);   // v_exp_f32 path
        const float a = __builtin_nontemporal_load(aRow + (size_t)v * n);
        const float r = acc[v] + cur_sim + 0.5f - sigma * (di + dj) - a;
        __builtin_nontemporal_store(r, oRow + (size_t)v * n);
    }
}

extern "C" void kernel_launch(void* const* d_in, const int* in_sizes, int n_in,
                              void* d_out, int out_size, void* d_ws, size_t ws_size,
                              hipStream_t stream) {
    const float* curvature = (const float*)d_in[0];
    const float* density   = (const float*)d_in[1];
    const float* normals   = (const float*)d_in[2];
    const float* linearity = (const float*)d_in[3];
    const float* aniso     = (const float*)d_in[4];
    const float* gamma_p   = (const float*)d_in[5];
    const float* sigma_p   = (const float*)d_in[6];
    const float* cscale_p  = (const float*)d_in[7];
    float* out = (float*)d_out;

    // in_sizes: [B*N, B*N, B*N*3, B*N, B*N*N, 1, 1, 1]
    const int bn = in_sizes[0];
    const int n  = in_sizes[4] / bn;   // N
    const int b  = bn / n;             // B

    dim3 block(256, 1, 1);
    dim3 grid((unsigned)(n / 16 / TILES_J_PER_WG), (unsigned)(n / 16), (unsigned)b);
    GeometryAttentionBias_kernel<<<grid, block, 0, stream>>>(
        curvature, density, normals, linearity, aniso,
        gamma_p, sigma_p, cscale_p, out, n);
}